// LengthRegulator_63230508531891
// MI455X (gfx1250) — compile-verified
//
#include <hip/hip_runtime.h>
#include <cstdint>

// Problem constants (from the reference): B=16, T_IN=512, D=768, dur in [0,8)
#define BATCH      16
#define T_IN       512
#define D_MODEL    768
#define CHUNKS     6            // 768 floats / (32 lanes * 4 floats) = 6 chunks of 512B
#define WAVES_PB   8            // waves (phonemes) per 256-thread block

// ---------------------------------------------------------------------------
// CDNA5 async global<->LDS data-mover path (ASYNCcnt-tracked).
// Builtins take AS(1)/AS(3) int4* (confirmed by round-1 diagnostics).
// ---------------------------------------------------------------------------
#if __has_builtin(__builtin_amdgcn_global_load_async_to_lds_b128) && \
    __has_builtin(__builtin_amdgcn_global_store_async_from_lds_b128) && \
    __has_builtin(__builtin_amdgcn_s_wait_asynccnt)
#define HAS_ASYNC 1
#else
#define HAS_ASYNC 0
#endif

#if HAS_ASYNC
typedef int v4i __attribute__((vector_size(16)));
typedef __attribute__((address_space(1))) v4i* g_v4i;   // global int4*
typedef __attribute__((address_space(3))) v4i* l_v4i;   // LDS int4*

__device__ __forceinline__ g_v4i g_cast(const void* p) {
  // AS(1) pointers share the 64-bit representation of generic global pointers.
  return (g_v4i)(uintptr_t)p;
}
__device__ __forceinline__ l_v4i lds_cast(void* p) {
  // Generic LDS pointer carries the LDS byte address in its low 32 bits.
  return (l_v4i)(unsigned)(uintptr_t)p;
}
#endif

// ---------------------------------------------------------------------------
// Emission probe (never launched): placed FIRST in the file so the disasm
// snippet shows the async mnemonics directly.
// ---------------------------------------------------------------------------
#if HAS_ASYNC
extern "C" __global__ void aa_async_emit_probe(const float* __restrict__ g,
                                               float* __restrict__ o) {
  __shared__ float l[128];
  float* lp = &l[(threadIdx.x & 31) * 4];
  __builtin_amdgcn_global_load_async_to_lds_b128(
      g_cast(g + (threadIdx.x & 31) * 4), lds_cast(lp), 0, 0);
  __builtin_amdgcn_s_wait_asynccnt(0);
  __builtin_amdgcn_global_store_async_from_lds_b128(
      g_cast(o + (threadIdx.x & 31) * 4), lds_cast(lp), 0, 0);
}
#endif

// ---------------------------------------------------------------------------
// Kernel B: one wave32 per phoneme. Stage the 3KB row in LDS once via async
// loads, then async-store it to each of its `dur` output frames.
// Control flow is wave-uniform; all addresses 16B-aligned.
// ---------------------------------------------------------------------------
extern "C" __global__ void lr_expand(const float* __restrict__ hidden,
                                     const int* __restrict__ cum,
                                     float* __restrict__ out,
                                     int T_out) {
#if HAS_ASYNC
  __shared__ float smem[WAVES_PB * D_MODEL];   // 24 KB / workgroup
#endif
  const int wave = threadIdx.x >> 5;
  const int lane = threadIdx.x & 31;
  const int p    = blockIdx.x * WAVES_PB + wave;   // phoneme id in [0, B*T_IN)
  const int b    = p >> 9;                         // / T_IN (T_IN = 512)
  const int i    = p & (T_IN - 1);

  const int end   = cum[p];
  const int start = (i == 0) ? 0 : cum[p - 1];
  if (end <= start) return;                        // zero-duration phoneme

  const float* src = hidden + (size_t)p * D_MODEL;

#if HAS_ASYNC
  float* lrow = &smem[wave * D_MODEL];
#pragma unroll
  for (int c = 0; c < CHUNKS; ++c) {
    __builtin_amdgcn_global_load_async_to_lds_b128(
        g_cast(src + c * 128 + lane * 4),
        lds_cast(lrow + c * 128 + lane * 4),
        /*offset=*/0, /*cpol=*/0);
  }
  __builtin_amdgcn_s_wait_asynccnt(0);             // row resident in LDS
  for (int f = start; f < end; ++f) {
    float* dst = out + ((size_t)b * T_out + f) * D_MODEL;
#pragma unroll
    for (int c = 0; c < CHUNKS; ++c) {
      __builtin_amdgcn_global_store_async_from_lds_b128(
          g_cast(dst + c * 128 + lane * 4),
          lds_cast(lrow + c * 128 + lane * 4),
          /*offset=*/0, /*cpol=*/0);
    }
  }
  // implicit S_WAIT_IDLE at S_ENDPGM drains outstanding async stores
#else
  float4 r[CHUNKS];
#pragma unroll
  for (int c = 0; c < CHUNKS; ++c)
    r[c] = *(const float4*)(src + c * 128 + lane * 4);
  for (int f = start; f < end; ++f) {
    float* dst = out + ((size_t)b * T_out + f) * D_MODEL;
#pragma unroll
    for (int c = 0; c < CHUNKS; ++c)
      *(float4*)(dst + c * 128 + lane * 4) = r[c];
  }
#endif
}

// ---------------------------------------------------------------------------
// Kernel A: per-batch inclusive prefix sum of durations (int64 -> int32).
// One 512-thread block per batch row; Hillis-Steele scan in LDS.
// ---------------------------------------------------------------------------
extern "C" __global__ void lr_scan(const long long* __restrict__ dur,
                                   int* __restrict__ cum,
                                   int* __restrict__ totals) {
  __shared__ int s[T_IN];
  const int b = blockIdx.x;
  const int i = threadIdx.x;
  s[i] = (int)dur[(size_t)b * T_IN + i];
  __syncthreads();
#pragma unroll
  for (int off = 1; off < T_IN; off <<= 1) {
    int t = (i >= off) ? s[i - off] : 0;
    __syncthreads();
    s[i] += t;
    __syncthreads();
  }
  cum[(size_t)b * T_IN + i] = s[i];
  if (i == T_IN - 1) totals[b] = s[i];
}

// ---------------------------------------------------------------------------
// Kernel C: zero only the tail frames [total[b], T_out) per batch.
// ---------------------------------------------------------------------------
extern "C" __global__ void lr_tail_zero(const int* __restrict__ totals,
                                        float* __restrict__ out,
                                        int T_out) {
  const int b     = blockIdx.x;
  const int total = totals[b];
  if (total >= T_out) return;
  float* base     = out + ((size_t)b * T_out + total) * D_MODEL;
  const size_t n4 = (size_t)(T_out - total) * (D_MODEL / 4);   // float4 count
  const float4 z  = make_float4(0.f, 0.f, 0.f, 0.f);
  for (size_t k = threadIdx.x; k < n4; k += blockDim.x)
    ((float4*)base)[k] = z;
}

// ---------------------------------------------------------------------------
// Host launcher
// ---------------------------------------------------------------------------
extern "C" void kernel_launch(void* const* d_in, const int* in_sizes, int n_in,
                              void* d_out, int out_size, void* d_ws, size_t ws_size,
                              hipStream_t stream) {
  const float*     hidden = (const float*)d_in[0];       // (B, T_IN, D) f32
  const long long* dur    = (const long long*)d_in[1];   // (B, T_IN) i64
  float*           out    = (float*)d_out;               // (B, T_out, D) f32

  const int T_out = out_size / (BATCH * D_MODEL);        // data-dependent, from harness

  int* cum    = (int*)d_ws;                              // B*T_IN ints
  int* totals = cum + BATCH * T_IN;                      // B ints

  lr_scan<<<BATCH, T_IN, 0, stream>>>(dur, cum, totals);

  const int nPhon = BATCH * T_IN;
  lr_expand<<<nPhon / WAVES_PB, WAVES_PB * 32, 0, stream>>>(hidden, cum, out, T_out);

  lr_tail_zero<<<BATCH, 256, 0, stream>>>(totals, out, T_out);
}